// Decoder_81647328297369
// MI455X (gfx1250) — compile-verified
//
#include <hip/hip_runtime.h>
#include <hip/hip_bf16.h>

// ---------------------------------------------------------------------------
// Decoder step for MI455X (gfx1250, wave32).
// Problem sizes (fixed): V=23262, E=300, H=1024, S=2048, B=64.
// Memory-bound on the two encoder_out streams (2 x 512MB); dense projections
// run on V_WMMA_F32_16X16X4_F32 (full f32 precision matrix pipe).
// ---------------------------------------------------------------------------

#define DV 23262
#define DE 300
#define DH 1024
#define DS 2048
#define DB 64
#define KLSTM (DH + DE)        // 1324
#define KCAT  (2 * DH)         // 2048
#define NCHUNK 16

typedef __attribute__((ext_vector_type(2))) float v2f;
typedef __attribute__((ext_vector_type(8))) float v8f;

__device__ __forceinline__ float sigmoidf_(float x) {
  return 1.0f / (1.0f + __expf(-x));
}

// ---------------------------------------------------------------------------
// Generic f32 WMMA GEMM:  Y[m,n] = (ACC ? Y : 0) + bias1[n] + bias2[n]
//                                  + sum_k X[m, k] * Welem(k, n)
//   W_KN = true  : Welem(k,n) = W[k*ldw + n]   (W stored [K, N])
//   W_KN = false : Welem(k,n) = W[n*ldw + k]   (W stored [N, K], i.e. X @ W^T)
// One wave per 16x16 D tile, 4 waves per block. M must be a multiple of 16.
// ---------------------------------------------------------------------------
template <bool W_KN, bool ACC, bool TANH>
__global__ __launch_bounds__(128) void gemm_wmma_f32(
    const float* __restrict__ X, int ldx,
    const float* __restrict__ W, int ldw,
    const float* __restrict__ b1, const float* __restrict__ b2,
    float* __restrict__ Y, int ldy, int M, int N, int K) {
  const int lane = threadIdx.x & 31;
  const int wv   = threadIdx.x >> 5;
  const int tn   = blockIdx.x * 4 + wv;
  const int n0   = tn * 16;
  const int m0   = blockIdx.y * 16;
  if (n0 >= N) return;                       // uniform per-wave: EXEC stays all-1 for WMMA

  const int half = lane >> 4;                // 0: K={0,1}, 1: K={2,3}
  const int l16  = lane & 15;
  const int koff = half * 2;
  const int m    = m0 + l16;                 // A row (always valid: M % 16 == 0)
  const int n    = n0 + l16;                 // B/D column
  const int nc   = (n < N) ? n : (N - 1);    // clamp for safe loads on tail tiles

  v8f acc;
#pragma unroll
  for (int r = 0; r < 8; ++r) {
    if (ACC) {
      const int mr = m0 + r + half * 8;
      acc[r] = (n < N) ? Y[(size_t)mr * ldy + n] : 0.0f;
    } else {
      acc[r] = 0.0f;
    }
  }

  const float* xrow = X + (size_t)m * ldx;
  for (int k0 = 0; k0 < K; k0 += 4) {
    // A 16x4 f32: lane holds X[m][k0+koff], X[m][k0+koff+1]
    v2f a = *(const v2f*)(xrow + k0 + koff);
    // B 4x16 f32: lane holds B[k0+koff][n], B[k0+koff+1][n]
    v2f b;
    if (W_KN) {
      b[0] = W[(size_t)(k0 + koff) * ldw + nc];
      b[1] = W[(size_t)(k0 + koff + 1) * ldw + nc];
    } else {
      b = *(const v2f*)(W + (size_t)nc * ldw + k0 + koff);
    }
    acc = __builtin_amdgcn_wmma_f32_16x16x4_f32(
        /*neg_a=*/false, a, /*neg_b=*/false, b,
        /*c_mod=*/(short)0, acc, /*reuse_a=*/false, /*reuse_b=*/false);
  }

  float bias = 0.0f;
  if (b1) bias += b1[nc];
  if (b2) bias += b2[nc];
  if (n < N) {
#pragma unroll
    for (int r = 0; r < 8; ++r) {
      const int mr = m0 + r + half * 8;
      float v = acc[r] + bias;
      if (TANH) v = tanhf(v);
      Y[(size_t)mr * ldy + n] = v;
    }
  }
}

// ---------------------------------------------------------------------------
// prep: embedding gather into lstm-concat cols [H, H+E); h0 copy into
// attention-concat cols [H, 2H).
// ---------------------------------------------------------------------------
__global__ __launch_bounds__(256) void prep_kernel(
    const int* __restrict__ tokens, const float* __restrict__ emb,
    const float* __restrict__ h0, float* __restrict__ xcat,
    float* __restrict__ xlstm) {
  const int tid = blockIdx.x * blockDim.x + threadIdx.x;
  if (tid < DB * DH) {
    const int b = tid / DH, h = tid % DH;
    xcat[(size_t)b * KCAT + DH + h] = h0[tid];
  }
  if (tid < DB * DE) {
    const int b = tid / DE, e = tid % DE;
    xlstm[(size_t)b * KLSTM + DH + e] = emb[(size_t)tokens[b] * DE + e];
  }
}

// ---------------------------------------------------------------------------
// scores[s,b] = enc[s,b,:] . q[b,:]   -> stored transposed as scoresT[b*S+s]
// One wave32 per (s,b) row; float4 coalesced; shfl_xor reduction.
// ---------------------------------------------------------------------------
__global__ __launch_bounds__(256) void scores_kernel(
    const float* __restrict__ enc, const float* __restrict__ q,
    float* __restrict__ scoresT) {
  const int wave = (blockIdx.x * blockDim.x + threadIdx.x) >> 5;
  const int lane = threadIdx.x & 31;
  if (wave >= DS * DB) return;
  const int s = wave / DB, b = wave % DB;
  const float4* row = (const float4*)(enc + ((size_t)s * DB + b) * DH);
  const float4* qv  = (const float4*)(q + (size_t)b * DH);
  float acc = 0.0f;
#pragma unroll
  for (int j = lane; j < DH / 4; j += 32) {
    const float4 e = row[j];
    const float4 w = qv[j];
    acc += e.x * w.x + e.y * w.y + e.z * w.z + e.w * w.w;
  }
#pragma unroll
  for (int m = 16; m >= 1; m >>= 1) acc += __shfl_xor(acc, m, 32);
  if (lane == 0) scoresT[(size_t)b * DS + s] = acc;
}

// ---------------------------------------------------------------------------
// In-place softmax over S for each batch column (one block per b).
// ---------------------------------------------------------------------------
__global__ __launch_bounds__(256) void softmax_kernel(float* __restrict__ scoresT) {
  const int b = blockIdx.x;
  float* p = scoresT + (size_t)b * DS;
  const int tid = threadIdx.x;
  __shared__ float red[8];

  float mx = -3.4e38f;
  for (int i = tid; i < DS; i += 256) mx = fmaxf(mx, p[i]);
#pragma unroll
  for (int m = 16; m >= 1; m >>= 1) mx = fmaxf(mx, __shfl_xor(mx, m, 32));
  if ((tid & 31) == 0) red[tid >> 5] = mx;
  __syncthreads();
  float bmax = red[0];
#pragma unroll
  for (int j = 1; j < 8; ++j) bmax = fmaxf(bmax, red[j]);
  __syncthreads();

  float sum = 0.0f;
  for (int i = tid; i < DS; i += 256) {
    const float e = __expf(p[i] - bmax);
    p[i] = e;
    sum += e;
  }
#pragma unroll
  for (int m = 16; m >= 1; m >>= 1) sum += __shfl_xor(sum, m, 32);
  if ((tid & 31) == 0) red[tid >> 5] = sum;
  __syncthreads();
  float bsum = 0.0f;
#pragma unroll
  for (int j = 0; j < 8; ++j) bsum += red[j];
  const float inv = 1.0f / bsum;
  for (int i = tid; i < DS; i += 256) p[i] *= inv;
}

// ---------------------------------------------------------------------------
// context partials: block (b, chunk) accumulates att-weighted enc rows over
// its S-chunk. 256 threads x float4 = one full 1024-wide H row per step.
// Fixed summation order -> bit-deterministic (no atomics).
// ---------------------------------------------------------------------------
__global__ __launch_bounds__(256) void context_partial_kernel(
    const float* __restrict__ enc, const float* __restrict__ attT,
    float* __restrict__ part) {
  const int b = blockIdx.x;
  const int chunk = blockIdx.y;
  const int per = DS / NCHUNK;
  const int s0 = chunk * per;
  const int tid = threadIdx.x;
  float ax = 0.f, ay = 0.f, az = 0.f, aw = 0.f;
  for (int s = s0; s < s0 + per; ++s) {
    const float a = attT[(size_t)b * DS + s];
    const float4 e = ((const float4*)(enc + ((size_t)s * DB + b) * DH))[tid];
    ax += a * e.x; ay += a * e.y; az += a * e.z; aw += a * e.w;
  }
  float4 o; o.x = ax; o.y = ay; o.z = az; o.w = aw;
  ((float4*)(part + ((size_t)chunk * DB + b) * DH))[tid] = o;
}

__global__ __launch_bounds__(256) void context_reduce_kernel(
    const float* __restrict__ part, float* __restrict__ xcat) {
  const int b = blockIdx.x;
  const int tid = threadIdx.x;
  float ax = 0.f, ay = 0.f, az = 0.f, aw = 0.f;
#pragma unroll
  for (int c = 0; c < NCHUNK; ++c) {
    const float4 e = ((const float4*)(part + ((size_t)c * DB + b) * DH))[tid];
    ax += e.x; ay += e.y; az += e.z; aw += e.w;
  }
  float4 o; o.x = ax; o.y = ay; o.z = az; o.w = aw;
  ((float4*)(xcat + (size_t)b * KCAT))[tid] = o;   // cols [0, H)
}

// ---------------------------------------------------------------------------
// LSTM cell elementwise (PyTorch gate order i,f,g,o). hN written into d_out
// (that region is then reused as the logits GEMM input).
// ---------------------------------------------------------------------------
__global__ __launch_bounds__(256) void lstm_kernel(
    const float* __restrict__ gates, const float* __restrict__ c0,
    float* __restrict__ outH, float* __restrict__ outC) {
  const int tid = blockIdx.x * blockDim.x + threadIdx.x;
  if (tid >= DB * DH) return;
  const int b = tid / DH, h = tid % DH;
  const float* g = gates + (size_t)b * 4 * DH;
  const float i = sigmoidf_(g[h]);
  const float f = sigmoidf_(g[DH + h]);
  const float gg = tanhf(g[2 * DH + h]);
  const float o = sigmoidf_(g[3 * DH + h]);
  const float c = f * c0[tid] + i * gg;
  outC[tid] = c;
  outH[tid] = o * tanhf(c);
}

// ---------------------------------------------------------------------------
extern "C" void kernel_launch(void* const* d_in, const int* in_sizes, int n_in,
                              void* d_out, int out_size, void* d_ws, size_t ws_size,
                              hipStream_t stream) {
  const int*   tokens = (const int*)d_in[0];
  const float* enc    = (const float*)d_in[1];
  const float* h0     = (const float*)d_in[2];
  const float* c0     = (const float*)d_in[3];
  const float* emb    = (const float*)d_in[4];
  const float* W_in   = (const float*)d_in[5];
  const float* W_out  = (const float*)d_in[6];
  const float* W_ih   = (const float*)d_in[7];
  const float* W_hh   = (const float*)d_in[8];
  const float* b_ih   = (const float*)d_in[9];
  const float* b_hh   = (const float*)d_in[10];
  const float* W_gen  = (const float*)d_in[11];
  const float* b_gen  = (const float*)d_in[12];

  float* out    = (float*)d_out;
  float* logits = out;                              // [B, V]
  float* outH   = out + (size_t)DB * DV;            // [B, H]
  float* outC   = outH + (size_t)DB * DH;           // [B, H]

  // workspace layout (floats)
  float* ws = (float*)d_ws;
  float* q       = ws;                                  // B*H      = 65536
  float* scoresT = q + (size_t)DB * DH;                 // B*S      = 131072
  float* part    = scoresT + (size_t)DB * DS;           // NCHUNK*B*H
  float* xcat    = part + (size_t)NCHUNK * DB * DH;     // B*2H     = 131072
  float* xlstm   = xcat + (size_t)DB * KCAT;            // B*(H+E)  = 84736
  float* gates   = xlstm + (size_t)DB * KLSTM;          // B*4H     = 262144

  // 1) embedding gather + h0 copy into concat buffers
  prep_kernel<<<(DB * DH + 255) / 256, 256, 0, stream>>>(tokens, emb, h0, xcat, xlstm);

  // 2) q = h0 @ W_in        [64,1024] x [1024,1024] (W stored [K,N])
  gemm_wmma_f32<true, false, false><<<dim3((DH / 16 + 3) / 4, DB / 16), 128, 0, stream>>>(
      h0, DH, W_in, DH, nullptr, nullptr, q, DH, DB, DH, DH);

  // 3) scores[s,b] = enc[s,b,:] . q[b,:]     (first 512MB encoder stream)
  scores_kernel<<<(DS * DB) / 8, 256, 0, stream>>>(enc, q, scoresT);

  // 4) softmax over S per batch column
  softmax_kernel<<<DB, 256, 0, stream>>>(scoresT);

  // 5) context = sum_s att[s,b] * enc[s,b,:] (second 512MB encoder stream)
  context_partial_kernel<<<dim3(DB, NCHUNK), 256, 0, stream>>>(enc, scoresT, part);
  context_reduce_kernel<<<DB, 256, 0, stream>>>(part, xcat);

  // 6) ctx_hat = tanh([context; h0] @ W_out^T) -> lstm-input cols [0, H)
  gemm_wmma_f32<false, false, true><<<dim3((DH / 16 + 3) / 4, DB / 16), 128, 0, stream>>>(
      xcat, KCAT, W_out, KCAT, nullptr, nullptr, xlstm, KLSTM, DB, DH, KCAT);

  // 7) gates = lstm_in @ W_ih^T + b_ih + b_hh;  gates += h0 @ W_hh^T
  gemm_wmma_f32<false, false, false><<<dim3((4 * DH / 16 + 3) / 4, DB / 16), 128, 0, stream>>>(
      xlstm, KLSTM, W_ih, KLSTM, b_ih, b_hh, gates, 4 * DH, DB, 4 * DH, KLSTM);
  gemm_wmma_f32<false, true, false><<<dim3((4 * DH / 16 + 3) / 4, DB / 16), 128, 0, stream>>>(
      h0, DH, W_hh, DH, nullptr, nullptr, gates, 4 * DH, DB, 4 * DH, DH);

  // 8) LSTM elementwise -> hN (into d_out, reused as GEMM input), cN
  lstm_kernel<<<(DB * DH + 255) / 256, 256, 0, stream>>>(gates, c0, outH, outC);

  // 9) logits = hN @ W_gen^T + b_gen   (N=23262, tail tiles guarded)
  gemm_wmma_f32<false, false, false><<<dim3(((DV + 15) / 16 + 3) / 4, DB / 16), 128, 0, stream>>>(
      outH, DH, W_gen, DH, b_gen, nullptr, logits, DV, DB, DV, DH);
}